// DuPPAM_89069031784598
// MI455X (gfx1250) — compile-verified
//
#include <hip/hip_runtime.h>
#include <math.h>

// ---------------------------------------------------------------------------
// DGCNN-style forward for MI455X (gfx1250, wave32, WMMA).
// All GEMM-shaped contractions (kNN gram, conv2/conv1, attention FCs) run on
// V_WMMA_F32_16X16X4_F32, 4 m-tiles per wave with a branch-free inner loop
// (clamped addresses, masked K-tail). fp32 kept throughout (BN/softmax
// precision; network is memory-bound at 23.3 TB/s so fp32 WMMA is right).
// Workspace requirement: ~107M floats (~426 MB).
// ---------------------------------------------------------------------------

#define BB  8
#define NN  1024
#define KNB 20
#define NKT (NN * KNB) // 20480

typedef __attribute__((ext_vector_type(2))) float v2f;
typedef __attribute__((ext_vector_type(8))) float v8f;

// ------------------------------ workspace map (float offsets) --------------
static const size_t OFF_PD   = 0;                       // 8,388,608  (B,N,N)
static const size_t OFF_IDX  = OFF_PD   + 8388608;      // 163,840 ints
static const size_t OFF_XX   = OFF_IDX  + 163840;       // 8,192
static const size_t OFF_Q    = OFF_XX   + 8192;         // (B,256,N)
static const size_t OFF_SRAW = OFF_Q    + 2097152;      // (B,N,K)
static const size_t OFF_ST   = OFF_SRAW + 163840;       // bn stats 2*2048
static const size_t OFF_XT   = OFF_ST   + 8192;         // (B,3,N)
static const size_t OFF_X2   = OFF_XT   + 24576;        // (B,64,N)
static const size_t OFF_X3   = OFF_X2   + 524288;
static const size_t OFF_X4   = OFF_X3   + 524288;       // (B,128,N)
static const size_t OFF_X5   = OFF_X4   + 1048576;      // (B,256,N)
static const size_t OFF_XC   = OFF_X5   + 2097152;      // (B,512,N)
static const size_t OFF_XG   = OFF_XC   + 4194304;      // (B,2048)
static const size_t OFF_G    = OFF_XG   + 16384;        // (B,2048)
static const size_t OFF_LF   = OFF_G    + 16384;        // (B,64)
static const size_t OFF_TMAX = OFF_LF   + 512;          // (B,128,N)
static const size_t OFF_HV   = OFF_TMAX + 1048576;      // (B,1024)
static const size_t OFF_Y1   = OFF_HV   + 8192;         // (512,8)
static const size_t OFF_Y2   = OFF_Y1   + 4096;         // (256,8)
static const size_t OFF_TV   = OFF_Y2   + 2048;         // (B,9)
static const size_t OFF_HA   = OFF_TV   + 128;          // (B,128,N)
static const size_t OFF_HB   = OFF_HA   + 1048576;      // (B,64,N)
static const size_t OFF_HC   = OFF_HB   + 524288;       // (B,64,N)
static const size_t OFF_SN   = OFF_HC   + 524288;       // (B,N)
static const size_t OFF_RA   = OFF_SN   + 8192;         // region A: 42M floats
static const size_t OFF_RB   = OFF_RA   + 42000000;     // region B: 42M floats
// total = OFF_RB + 42,000,000 = 106,444,672 floats (~426 MB)

// ------------------------------ kernels ------------------------------------

// Generic WMMA GEMM: Y[b,o,m] = sum_c W[o,c] * X[b,c,m]  (xrm: X is (m,c) rows)
// One wave computes a 16(o) x 64(m) slab: 4 accumulators share one A fragment.
// Main K-loop is branch-free: OOB rows/cols use clamped (valid, finite)
// addresses and are simply never stored. The K-tail (C%4) is peeled and
// masked on BOTH operands (avoid 0*Inf -> NaN in WMMA).
__global__ void k_gemm(const float* __restrict__ W, const float* __restrict__ X,
                       float* __restrict__ Y, int Bn, int O, int C, int M, int xrm)
{
    int wid  = (int)(((long)blockIdx.x * blockDim.x + threadIdx.x) >> 5);
    int lane = threadIdx.x & 31;
    int otiles  = (O + 15) >> 4;
    int mgroups = (M + 63) >> 6;
    long perB = (long)otiles * mgroups;
    if (wid >= (long)Bn * perB) return;
    int b  = (int)(wid / perB);
    int t  = (int)(wid % perB);
    int ot = t / mgroups, mg = t % mgroups;
    int row = lane & 15;
    int kh  = (lane >> 4) << 1;           // 0 or 2 (K half, f32 A/B layout)
    int oA  = (ot << 4) + row;
    const float* Wr = W + (size_t)(oA < O ? oA : O - 1) * C;
    int m0  = mg << 6;
    int mB0 = m0 + row,      mB1 = m0 + 16 + row;
    int mB2 = m0 + 32 + row, mB3 = m0 + 48 + row;
    int mc0 = mB0 < M ? mB0 : M - 1;
    int mc1 = mB1 < M ? mB1 : M - 1;
    int mc2 = mB2 < M ? mB2 : M - 1;
    int mc3 = mB3 < M ? mB3 : M - 1;
    const float* Xb = X + (size_t)b * C * M;
    v8f a0 = {0.f,0.f,0.f,0.f,0.f,0.f,0.f,0.f};
    v8f a1 = a0, a2 = a0, a3 = a0;
    int Cm = C & ~3;
    for (int k0 = 0; k0 < Cm; k0 += 4) {
        int c0 = k0 + kh;
        v2f af, b0, b1, b2, b3;
        af.x = Wr[c0]; af.y = Wr[c0 + 1];
        if (xrm) {
            const float* x0 = Xb + (size_t)mc0 * C + c0;
            const float* x1 = Xb + (size_t)mc1 * C + c0;
            const float* x2 = Xb + (size_t)mc2 * C + c0;
            const float* x3 = Xb + (size_t)mc3 * C + c0;
            b0.x = x0[0]; b0.y = x0[1];
            b1.x = x1[0]; b1.y = x1[1];
            b2.x = x2[0]; b2.y = x2[1];
            b3.x = x3[0]; b3.y = x3[1];
        } else {
            const float* xr0 = Xb + (size_t)c0 * M;
            const float* xr1 = xr0 + M;
            b0.x = xr0[mc0]; b0.y = xr1[mc0];
            b1.x = xr0[mc1]; b1.y = xr1[mc1];
            b2.x = xr0[mc2]; b2.y = xr1[mc2];
            b3.x = xr0[mc3]; b3.y = xr1[mc3];
        }
        a0 = __builtin_amdgcn_wmma_f32_16x16x4_f32(false, af, false, b0, (short)0, a0, false, false);
        a1 = __builtin_amdgcn_wmma_f32_16x16x4_f32(false, af, false, b1, (short)0, a1, false, false);
        a2 = __builtin_amdgcn_wmma_f32_16x16x4_f32(false, af, false, b2, (short)0, a2, false, false);
        a3 = __builtin_amdgcn_wmma_f32_16x16x4_f32(false, af, false, b3, (short)0, a3, false, false);
    }
    if (Cm < C) {                          // masked tail chunk (C % 4 != 0)
        int c0 = Cm + kh, c1 = c0 + 1;
        int c0c = c0 < C ? c0 : 0, c1c = c1 < C ? c1 : 0;
        float f0 = c0 < C ? 1.f : 0.f, f1 = c1 < C ? 1.f : 0.f;
        v2f af, b0, b1, b2, b3;
        af.x = Wr[c0c] * f0; af.y = Wr[c1c] * f1;
        if (xrm) {
            b0.x = Xb[(size_t)mc0 * C + c0c] * f0; b0.y = Xb[(size_t)mc0 * C + c1c] * f1;
            b1.x = Xb[(size_t)mc1 * C + c0c] * f0; b1.y = Xb[(size_t)mc1 * C + c1c] * f1;
            b2.x = Xb[(size_t)mc2 * C + c0c] * f0; b2.y = Xb[(size_t)mc2 * C + c1c] * f1;
            b3.x = Xb[(size_t)mc3 * C + c0c] * f0; b3.y = Xb[(size_t)mc3 * C + c1c] * f1;
        } else {
            const float* xr0 = Xb + (size_t)c0c * M;
            const float* xr1 = Xb + (size_t)c1c * M;
            b0.x = xr0[mc0] * f0; b0.y = xr1[mc0] * f1;
            b1.x = xr0[mc1] * f0; b1.y = xr1[mc1] * f1;
            b2.x = xr0[mc2] * f0; b2.y = xr1[mc2] * f1;
            b3.x = xr0[mc3] * f0; b3.y = xr1[mc3] * f1;
        }
        a0 = __builtin_amdgcn_wmma_f32_16x16x4_f32(false, af, false, b0, (short)0, a0, false, false);
        a1 = __builtin_amdgcn_wmma_f32_16x16x4_f32(false, af, false, b1, (short)0, a1, false, false);
        a2 = __builtin_amdgcn_wmma_f32_16x16x4_f32(false, af, false, b2, (short)0, a2, false, false);
        a3 = __builtin_amdgcn_wmma_f32_16x16x4_f32(false, af, false, b3, (short)0, a3, false, false);
    }
    float* Yb = Y + (size_t)b * O * M;
    int orow = (ot << 4) + ((lane >> 4) << 3);
#pragma unroll
    for (int j = 0; j < 8; ++j) {
        int o = orow + j;
        if (o < O) {
            float* yr = Yb + (size_t)o * M;
            if (mB0 < M) yr[mB0] = a0[j];
            if (mB1 < M) yr[mB1] = a1[j];
            if (mB2 < M) yr[mB2] = a2[j];
            if (mB3 < M) yr[mB3] = a3[j];
        }
    }
}

// Gram + distance epilogue: pd[b,i,j] = 2*sum_c x[b,c,i]x[b,c,j] - xx[i] - xx[j]
// One wave computes 16(i) x 64(j); N=1024 -> no row/col bounds at all.
__global__ void k_gram(const float* __restrict__ X, const float* __restrict__ xx,
                       float* __restrict__ pd, int C)
{
    int wid  = (int)(((long)blockIdx.x * blockDim.x + threadIdx.x) >> 5);
    int lane = threadIdx.x & 31;
    const int jgroups = NN / 64;
    long perB = (long)(NN / 16) * jgroups;
    if (wid >= (long)BB * perB) return;
    int b  = (int)(wid / perB);
    int t  = (int)(wid % perB);
    int it = t / jgroups, jg = t % jgroups;
    int row = lane & 15;
    int kh  = (lane >> 4) << 1;
    int iA  = (it << 4) + row;
    int j0  = jg << 6;
    int jB0 = j0 + row,      jB1 = j0 + 16 + row;
    int jB2 = j0 + 32 + row, jB3 = j0 + 48 + row;
    const float* Xb = X + (size_t)b * C * NN;
    v8f a0 = {0.f,0.f,0.f,0.f,0.f,0.f,0.f,0.f};
    v8f a1 = a0, a2 = a0, a3 = a0;
    int Cm = C & ~3;
    for (int k0 = 0; k0 < Cm; k0 += 4) {
        int c0 = k0 + kh;
        const float* xr0 = Xb + (size_t)c0 * NN;
        const float* xr1 = xr0 + NN;
        v2f af, b0, b1, b2, b3;
        af.x = xr0[iA];  af.y = xr1[iA];
        b0.x = xr0[jB0]; b0.y = xr1[jB0];
        b1.x = xr0[jB1]; b1.y = xr1[jB1];
        b2.x = xr0[jB2]; b2.y = xr1[jB2];
        b3.x = xr0[jB3]; b3.y = xr1[jB3];
        a0 = __builtin_amdgcn_wmma_f32_16x16x4_f32(false, af, false, b0, (short)0, a0, false, false);
        a1 = __builtin_amdgcn_wmma_f32_16x16x4_f32(false, af, false, b1, (short)0, a1, false, false);
        a2 = __builtin_amdgcn_wmma_f32_16x16x4_f32(false, af, false, b2, (short)0, a2, false, false);
        a3 = __builtin_amdgcn_wmma_f32_16x16x4_f32(false, af, false, b3, (short)0, a3, false, false);
    }
    if (Cm < C) {                          // masked tail (C=3 case)
        int c0 = Cm + kh, c1 = c0 + 1;
        int c0c = c0 < C ? c0 : 0, c1c = c1 < C ? c1 : 0;
        float f0 = c0 < C ? 1.f : 0.f, f1 = c1 < C ? 1.f : 0.f;
        const float* xr0 = Xb + (size_t)c0c * NN;
        const float* xr1 = Xb + (size_t)c1c * NN;
        v2f af, b0, b1, b2, b3;
        af.x = xr0[iA]  * f0; af.y = xr1[iA]  * f1;
        b0.x = xr0[jB0] * f0; b0.y = xr1[jB0] * f1;
        b1.x = xr0[jB1] * f0; b1.y = xr1[jB1] * f1;
        b2.x = xr0[jB2] * f0; b2.y = xr1[jB2] * f1;
        b3.x = xr0[jB3] * f0; b3.y = xr1[jB3] * f1;
        a0 = __builtin_amdgcn_wmma_f32_16x16x4_f32(false, af, false, b0, (short)0, a0, false, false);
        a1 = __builtin_amdgcn_wmma_f32_16x16x4_f32(false, af, false, b1, (short)0, a1, false, false);
        a2 = __builtin_amdgcn_wmma_f32_16x16x4_f32(false, af, false, b2, (short)0, a2, false, false);
        a3 = __builtin_amdgcn_wmma_f32_16x16x4_f32(false, af, false, b3, (short)0, a3, false, false);
    }
    const float* xxb = xx + (size_t)b * NN;
    float xj0 = xxb[jB0], xj1 = xxb[jB1], xj2 = xxb[jB2], xj3 = xxb[jB3];
    float* pdb = pd + (size_t)b * NN * NN;
    int ir0 = (it << 4) + ((lane >> 4) << 3);
#pragma unroll
    for (int j = 0; j < 8; ++j) {
        int i = ir0 + j;
        float xi = xxb[i];
        float* pr = pdb + (size_t)i * NN;
        pr[jB0] = 2.f * a0[j] - xi - xj0;
        pr[jB1] = 2.f * a1[j] - xi - xj1;
        pr[jB2] = 2.f * a2[j] - xi - xj2;
        pr[jB3] = 2.f * a3[j] - xi - xj3;
    }
}

__global__ void k_sqnorm(const float* __restrict__ X, float* __restrict__ xx, int C)
{
    int t = blockIdx.x * 256 + threadIdx.x;
    if (t >= BB * NN) return;
    int b = t / NN, n = t % NN;
    const float* Xb = X + (size_t)b * C * NN + n;
    float s = 0.f;
    for (int c = 0; c < C; ++c) { float v = Xb[(size_t)c * NN]; s += v * v; }
    xx[t] = s;
}

// top-20 (largest pd == nearest): one wave per row; insertion list + tournament
__global__ void k_topk(const float* __restrict__ pd, int* __restrict__ idx)
{
    int r = blockIdx.x;                 // b*N + n
    int lane = threadIdx.x;
    const float* p = pd + (size_t)r * NN;
    float vals[KNB]; int ids[KNB];
#pragma unroll
    for (int i = 0; i < KNB; ++i) { vals[i] = -3.4e38f; ids[i] = 0x7fffffff; }
    for (int j = lane; j < NN; j += 32) {
        float v = p[j];
        if (v > vals[KNB-1] || (v == vals[KNB-1] && j < ids[KNB-1])) {
            int pos = KNB - 1;
            while (pos > 0 && (v > vals[pos-1] ||
                              (v == vals[pos-1] && j < ids[pos-1]))) {
                vals[pos] = vals[pos-1]; ids[pos] = ids[pos-1]; --pos;
            }
            vals[pos] = v; ids[pos] = j;
        }
    }
    int head = 0;
    for (int t = 0; t < KNB; ++t) {
        float v  = (head < KNB) ? vals[head] : -3.4e38f;
        int   id = (head < KNB) ? ids[head]  : 0x7fffffff;
        float bv = v; int bi = id;
        for (int off = 16; off; off >>= 1) {
            float ov = __shfl_xor(bv, off);
            int   oi = __shfl_xor(bi, off);
            if (ov > bv || (ov == bv && oi < bi)) { bv = ov; bi = oi; }
        }
        if (v == bv && id == bi) ++head;
        if (lane == 0) idx[(size_t)r * KNB + t] = bi;
    }
}

// build edge features: normal: [nb-ctr, ctr]; first: [|d|^2, d, ctr, nb] (C=3)
__global__ void k_feat(const float* __restrict__ X, const int* __restrict__ idx,
                       float* __restrict__ feat, int C, int first)
{
    int t = blockIdx.x * 256 + threadIdx.x;
    if (t >= BB * NN * KNB) return;
    int k = t % KNB; int n = (t / KNB) % NN; int b = t / (KNB * NN);
    int j = idx[t];
    const float* Xb = X + (size_t)b * C * NN;
    const size_t MM = (size_t)NN * KNB;
    if (first) {
        float* fb = feat + (size_t)b * 10 * MM + (size_t)n * KNB + k;
        float d2 = 0.f;
#pragma unroll
        for (int c = 0; c < 3; ++c) {
            float nv = Xb[(size_t)c * NN + j];
            float cv = Xb[(size_t)c * NN + n];
            float d = nv - cv; d2 += d * d;
            fb[(size_t)(1 + c) * MM] = d;
            fb[(size_t)(4 + c) * MM] = cv;
            fb[(size_t)(7 + c) * MM] = nv;
        }
        fb[0] = d2;
    } else {
        float* fb = feat + (size_t)b * (2 * C) * MM + (size_t)n * KNB + k;
        for (int c = 0; c < C; ++c) {
            float nv = Xb[(size_t)c * NN + j];
            float cv = Xb[(size_t)c * NN + n];
            fb[(size_t)c * MM]       = nv - cv;
            fb[(size_t)(C + c) * MM] = cv;
        }
    }
}

// per-channel sum/sumsq over (b, m); layout Y (Bn, O, M)
__global__ void k_bn_stats(const float* __restrict__ Y, float* __restrict__ st,
                           int O, long M)
{
    __shared__ float s1[256], s2[256];
    int o = blockIdx.x, b = blockIdx.y;
    const float* p = Y + ((size_t)b * O + o) * M;
    float a = 0.f, q = 0.f;
    for (long i = threadIdx.x; i < M; i += 256) { float v = p[i]; a += v; q += v * v; }
    s1[threadIdx.x] = a; s2[threadIdx.x] = q; __syncthreads();
    for (int s = 128; s; s >>= 1) {
        if ((int)threadIdx.x < s) {
            s1[threadIdx.x] += s1[threadIdx.x + s];
            s2[threadIdx.x] += s2[threadIdx.x + s];
        }
        __syncthreads();
    }
    if (threadIdx.x == 0) { atomicAdd(&st[o], s1[0]); atomicAdd(&st[O + o], s2[0]); }
}

__global__ void k_bn_apply(float* __restrict__ Y, const float* __restrict__ st,
                           int O, long M, long total, float invc, int relu)
{
    long t = (long)blockIdx.x * 256 + threadIdx.x;
    if (t >= total) return;
    int o = (int)((t / M) % O);
    float mean = st[o] * invc;
    float var  = st[O + o] * invc - mean * mean;
    float v = (Y[t] - mean) * rsqrtf(var + 1e-5f);
    if (relu && v < 0.f) v *= 0.2f;
    Y[t] = v;
}

// sraw[b,n,k] = sum_o Ws[o] * (q[b,o,n] + v[b,o,n,k])
__global__ void k_sconv(const float* __restrict__ q, const float* __restrict__ v,
                        const float* __restrict__ Ws, float* __restrict__ sraw, int O)
{
    int t = blockIdx.x * 256 + threadIdx.x;
    if (t >= BB * NN * KNB) return;
    int k = t % KNB; int n = (t / KNB) % NN; int b = t / (KNB * NN);
    size_t qbase = (size_t)b * O * NN + n;
    size_t vbase = ((size_t)b * O * NN) * KNB + (size_t)n * KNB + k;
    float s = 0.f;
    for (int o = 0; o < O; ++o)
        s += Ws[o] * (q[qbase + (size_t)o * NN] + v[vbase + (size_t)o * NN * KNB]);
    sraw[t] = s;
}

// softmax over K then weighted sum of v -> out (B,O,N); one wave per (b,n)
__global__ void k_lab_finish(const float* __restrict__ v, const float* __restrict__ s,
                             float* __restrict__ out, int O)
{
    int wid  = (int)(((long)blockIdx.x * blockDim.x + threadIdx.x) >> 5);
    int lane = threadIdx.x & 31;
    if (wid >= BB * NN) return;
    int b = wid / NN, n = wid % NN;
    const float* sp = s + (size_t)wid * KNB;
    float sv = (lane < KNB) ? sp[lane] : -3.4e38f;
    float m = sv;
    for (int off = 16; off; off >>= 1) m = fmaxf(m, __shfl_xor(m, off));
    float e = (lane < KNB) ? __expf(sv - m) : 0.f;
    float sum = e;
    for (int off = 16; off; off >>= 1) sum += __shfl_xor(sum, off);
    float w = e / sum;
    float wk[KNB];
#pragma unroll
    for (int kk = 0; kk < KNB; ++kk) wk[kk] = __shfl(w, kk);
    for (int o = lane; o < O; o += 32) {
        const float* vp = v + ((size_t)b * O + o) * ((size_t)NN * KNB) + (size_t)n * KNB;
        float acc = 0.f;
#pragma unroll
        for (int kk = 0; kk < KNB; ++kk) acc += vp[kk] * wk[kk];
        out[((size_t)b * O + o) * NN + n] = acc;
    }
}

__global__ void k_maxk(const float* __restrict__ in, float* __restrict__ out, int O)
{
    int t = blockIdx.x * 256 + threadIdx.x;
    if (t >= BB * O * NN) return;
    const float* p = in + (size_t)t * KNB;
    float m = p[0];
#pragma unroll
    for (int k = 1; k < KNB; ++k) m = fmaxf(m, p[k]);
    out[t] = m;
}

__global__ void k_maxlast(const float* __restrict__ in, float* __restrict__ out,
                          int rows, int L)
{
    int wid  = (int)(((long)blockIdx.x * blockDim.x + threadIdx.x) >> 5);
    int lane = threadIdx.x & 31;
    if (wid >= rows) return;
    const float* p = in + (size_t)wid * L;
    float m = -3.4e38f;
    for (int i = lane; i < L; i += 32) m = fmaxf(m, p[i]);
    for (int off = 16; off; off >>= 1) m = fmaxf(m, __shfl_xor(m, off));
    if (!lane) out[wid] = m;
}

// t[b,:9] = y2col(b) @ tt_w.T + tt_b   (y2 is (256, 8) channel-major)
__global__ void k_fc_small(const float* __restrict__ y2, const float* __restrict__ Wt,
                           const float* __restrict__ bt, float* __restrict__ tv)
{
    int t = blockIdx.x * 64 + threadIdx.x;
    if (t >= BB * 9) return;
    int b = t / 9, o = t % 9;
    float s = bt[o];
    for (int c = 0; c < 256; ++c) s += y2[(size_t)c * BB + b] * Wt[(size_t)o * 256 + c];
    tv[t] = s;
}

__global__ void k_transform_apply(const float* __restrict__ x, const float* __restrict__ tv,
                                  float* __restrict__ xo)
{
    int t = blockIdx.x * 256 + threadIdx.x;
    if (t >= BB * 3 * NN) return;
    int n = t % NN; int d = (t / NN) % 3; int b = t / (3 * NN);
    float s = 0.f;
#pragma unroll
    for (int c = 0; c < 3; ++c)
        s += x[((size_t)b * 3 + c) * NN + n] * tv[b * 9 + c * 3 + d];
    xo[t] = s;
}

// x_glo[b, o]: o<N -> sum_c x ; else mean_c x at point (o-N)
__global__ void k_xglo(const float* __restrict__ x, float* __restrict__ g)
{
    int t = blockIdx.x * 256 + threadIdx.x;
    if (t >= BB * 2 * NN) return;
    int o = t % (2 * NN); int b = t / (2 * NN);
    int n = (o < NN) ? o : o - NN;
    float s = x[((size_t)b * 3 + 0) * NN + n] + x[((size_t)b * 3 + 1) * NN + n]
            + x[((size_t)b * 3 + 2) * NN + n];
    g[t] = (o < NN) ? s : s * (1.f / 3.f);
}

__global__ void k_concat_xc(const float* __restrict__ x2, const float* __restrict__ x3,
                            const float* __restrict__ x4, const float* __restrict__ x5,
                            float* __restrict__ xc)
{
    long t = (long)blockIdx.x * 256 + threadIdx.x;
    if (t >= (long)BB * 512 * NN) return;
    int n = (int)(t % NN); int c = (int)((t / NN) % 512); int b = (int)(t / ((long)512 * NN));
    float v;
    if      (c < 64)  v = x2[((size_t)b * 64  + c)         * NN + n];
    else if (c < 128) v = x3[((size_t)b * 64  + (c - 64))  * NN + n];
    else if (c < 256) v = x4[((size_t)b * 128 + (c - 128)) * NN + n];
    else              v = x5[((size_t)b * 256 + (c - 256)) * NN + n];
    xc[t] = v;
}

__global__ void k_add_bcast(const float* __restrict__ vals, const float* __restrict__ g,
                            float* __restrict__ f)
{
    long t = (long)blockIdx.x * 256 + threadIdx.x;
    if (t >= (long)BB * 2048 * NN) return;
    int o = (int)((t / NN) % 2048); int b = (int)(t / ((long)2048 * NN));
    f[t] = vals[t] + g[b * 2048 + o];
}

__global__ void k_softmax_n(float* __restrict__ sn)
{
    __shared__ float red[256];
    int b = blockIdx.x, t = threadIdx.x;
    float* p = sn + (size_t)b * NN;
    float m = -3.4e38f;
    for (int i = t; i < NN; i += 256) m = fmaxf(m, p[i]);
    red[t] = m; __syncthreads();
    for (int s = 128; s; s >>= 1) { if (t < s) red[t] = fmaxf(red[t], red[t + s]); __syncthreads(); }
    m = red[0]; __syncthreads();
    float sum = 0.f;
    for (int i = t; i < NN; i += 256) { float e = __expf(p[i] - m); p[i] = e; sum += e; }
    red[t] = sum; __syncthreads();
    for (int s = 128; s; s >>= 1) { if (t < s) red[t] += red[t + s]; __syncthreads(); }
    float inv = 1.f / red[0];
    for (int i = t; i < NN; i += 256) p[i] *= inv;
}

__global__ void k_gab_gather(const float* __restrict__ vals, const float* __restrict__ s,
                             float* __restrict__ g)
{
    int wid  = (int)(((long)blockIdx.x * blockDim.x + threadIdx.x) >> 5);
    int lane = threadIdx.x & 31;
    if (wid >= BB * 2048) return;
    int b = wid / 2048;
    const float* vp = vals + (size_t)wid * NN;
    const float* sp = s + (size_t)b * NN;
    float acc = 0.f;
    for (int n = lane; n < NN; n += 32) acc += vp[n] * sp[n];
    for (int off = 16; off; off >>= 1) acc += __shfl_xor(acc, off);
    if (!lane) g[wid] = acc;
}

__global__ void k_concat_h(const float* __restrict__ g, const float* __restrict__ lf,
                           const float* __restrict__ x2, const float* __restrict__ x3,
                           const float* __restrict__ x4, const float* __restrict__ x5,
                           float* __restrict__ h)
{
    long t = (long)blockIdx.x * 256 + threadIdx.x;
    if (t >= (long)BB * 2624 * NN) return;
    int n = (int)(t % NN); int c = (int)((t / NN) % 2624); int b = (int)(t / ((long)2624 * NN));
    float v;
    if      (c < 2048) v = g[b * 2048 + c];
    else if (c < 2112) v = lf[b * 64 + (c - 2048)];
    else if (c < 2176) v = x2[((size_t)b * 64  + (c - 2112)) * NN + n];
    else if (c < 2240) v = x3[((size_t)b * 64  + (c - 2176)) * NN + n];
    else if (c < 2368) v = x4[((size_t)b * 128 + (c - 2240)) * NN + n];
    else               v = x5[((size_t)b * 256 + (c - 2368)) * NN + n];
    h[t] = v;
}

// ------------------------------ host helpers -------------------------------
static inline int cdiv(long a, long b) { return (int)((a + b - 1) / b); }

static void gemm(const float* W, const float* X, float* Y,
                 int Bn, int O, int C, int M, int xrm, hipStream_t s)
{
    long waves = (long)Bn * ((O + 15) / 16) * ((M + 63) / 64);
    k_gemm<<<cdiv(waves * 32, 256), 256, 0, s>>>(W, X, Y, Bn, O, C, M, xrm);
}

static void bn(float* Y, float* st, int Bn, int O, long M, int relu, hipStream_t s)
{
    hipMemsetAsync(st, 0, sizeof(float) * 2 * (size_t)O, s);
    dim3 g((unsigned)O, (unsigned)Bn);
    k_bn_stats<<<g, 256, 0, s>>>(Y, st, O, M);
    long tot = (long)Bn * O * M;
    k_bn_apply<<<cdiv(tot, 256), 256, 0, s>>>(Y, st, O, M, tot,
                                              1.f / (float)((long)Bn * M), relu);
}

static void knn(const float* X, int C, float* wsf, hipStream_t s)
{
    float* pd = wsf + OFF_PD;
    float* xx = wsf + OFF_XX;
    int*   id = (int*)(wsf + OFF_IDX);
    k_sqnorm<<<cdiv((long)BB * NN, 256), 256, 0, s>>>(X, xx, C);
    long waves = (long)BB * (NN / 16) * (NN / 64);
    k_gram<<<cdiv(waves * 32, 256), 256, 0, s>>>(X, xx, pd, C);
    k_topk<<<BB * NN, 32, 0, s>>>(pd, id);
}

static void lab(const float* xin, int Cin, int first,
                const float* Wk, const float* Wv, const float* Ws, int O,
                float* out, float* wsf, hipStream_t s)
{
    int*   idx  = (int*)(wsf + OFF_IDX);
    float* feat = wsf + OFF_RA;
    float* v    = wsf + OFF_RB;
    float* q    = wsf + OFF_Q;
    float* sraw = wsf + OFF_SRAW;
    float* st   = wsf + OFF_ST;
    knn(xin, Cin, wsf, s);
    k_feat<<<cdiv((long)BB * NN * KNB, 256), 256, 0, s>>>(xin, idx, feat, Cin, first);
    int F = first ? 10 : 2 * Cin;
    gemm(Wk, xin, q, BB, O, Cin, NN, 0, s);  bn(q, st, BB, O, NN, 1, s);
    gemm(Wv, feat, v, BB, O, F, NKT, 0, s);  bn(v, st, BB, O, NKT, 1, s);
    k_sconv<<<cdiv((long)BB * NN * KNB, 256), 256, 0, s>>>(q, v, Ws, sraw, O);
    bn(sraw, st, 1, 1, (long)BB * NN * KNB, 1, s);
    k_lab_finish<<<cdiv((long)BB * NN * 32, 256), 256, 0, s>>>(v, sraw, out, O);
}

// ------------------------------ entry --------------------------------------
extern "C" void kernel_launch(void* const* d_in, const int* in_sizes, int n_in,
                              void* d_out, int out_size, void* d_ws, size_t ws_size,
                              hipStream_t stream)
{
    (void)in_sizes; (void)n_in; (void)out_size; (void)ws_size;
    // inputs in setup_inputs() dict (insertion) order: x, l, k, then params
    const float* x    = (const float*)d_in[0];
    const float* l    = (const float*)d_in[1];
    const float* tc1  = (const float*)d_in[3];
    const float* tc2  = (const float*)d_in[4];
    const float* tc3  = (const float*)d_in[5];
    const float* tl1  = (const float*)d_in[6];
    const float* tl2  = (const float*)d_in[7];
    const float* ttw  = (const float*)d_in[8];
    const float* l1k  = (const float*)d_in[9];
    const float* l1v  = (const float*)d_in[10];
    const float* l1s  = (const float*)d_in[11];
    const float* l2k  = (const float*)d_in[12];
    const float* l2v  = (const float*)d_in[13];
    const float* l2s  = (const float*)d_in[14];
    const float* l3k  = (const float*)d_in[15];
    const float* l3v  = (const float*)d_in[16];
    const float* l3s  = (const float*)d_in[17];
    const float* l4k  = (const float*)d_in[18];
    const float* l4v  = (const float*)d_in[19];
    const float* l4s  = (const float*)d_in[20];
    const float* g1   = (const float*)d_in[21];
    const float* g2   = (const float*)d_in[22];
    const float* g3   = (const float*)d_in[23];
    const float* g4   = (const float*)d_in[24];
    const float* c7w  = (const float*)d_in[25];
    const float* c8w  = (const float*)d_in[26];
    const float* c9w  = (const float*)d_in[27];
    const float* c10w = (const float*)d_in[28];
    const float* l5k  = (const float*)d_in[29];
    const float* l5v  = (const float*)d_in[30];
    const float* l5s  = (const float*)d_in[31];
    const float* l6k  = (const float*)d_in[32];
    const float* l6v  = (const float*)d_in[33];
    const float* l6s  = (const float*)d_in[34];
    const float* l7k  = (const float*)d_in[35];
    const float* l7v  = (const float*)d_in[36];
    const float* l7s  = (const float*)d_in[37];
    const float* ttb  = (const float*)d_in[38];

    float* wsf = (float*)d_ws;
    float* st   = wsf + OFF_ST;
    int*   idx  = (int*)(wsf + OFF_IDX);
    float* RA   = wsf + OFF_RA;
    float* RB   = wsf + OFF_RB;
    float* xt   = wsf + OFF_XT;
    float* x2   = wsf + OFF_X2;
    float* x3   = wsf + OFF_X3;
    float* x4   = wsf + OFF_X4;
    float* x5   = wsf + OFF_X5;
    float* xc   = wsf + OFF_XC;
    float* xg   = wsf + OFF_XG;
    float* gbuf = wsf + OFF_G;
    float* lf   = wsf + OFF_LF;
    float* tmax = wsf + OFF_TMAX;
    float* hv   = wsf + OFF_HV;
    float* y1   = wsf + OFF_Y1;
    float* y2   = wsf + OFF_Y2;
    float* tv   = wsf + OFF_TV;
    float* hA   = wsf + OFF_HA;
    float* hB   = wsf + OFF_HB;
    float* hC   = wsf + OFF_HC;
    float* sn   = wsf + OFF_SN;

    // --- transform_net on fea1 = graph_feature(x) (C=3 -> 6 channels) ---
    knn(x, 3, wsf, stream);
    k_feat<<<cdiv((long)BB * NN * KNB, 256), 256, 0, stream>>>(x, idx, RA, 3, 0);
    gemm(tc1, RA, RB, BB, 64, 6, NKT, 0, stream);   bn(RB, st, BB, 64, NKT, 1, stream);
    gemm(tc2, RB, RA, BB, 128, 64, NKT, 0, stream); bn(RA, st, BB, 128, NKT, 1, stream);
    k_maxk<<<cdiv((long)BB * 128 * NN, 256), 256, 0, stream>>>(RA, tmax, 128);
    gemm(tc3, tmax, RB, BB, 1024, 128, NN, 0, stream); bn(RB, st, BB, 1024, NN, 1, stream);
    k_maxlast<<<cdiv((long)BB * 1024 * 32, 256), 256, 0, stream>>>(RB, hv, BB * 1024, NN);
    gemm(tl1, hv, y1, 1, 512, 1024, BB, 1, stream); bn(y1, st, 1, 512, BB, 1, stream);
    gemm(tl2, y1, y2, 1, 256, 512, BB, 0, stream);  bn(y2, st, 1, 256, BB, 1, stream);
    k_fc_small<<<cdiv(BB * 9, 64), 64, 0, stream>>>(y2, ttw, ttb, tv);
    k_transform_apply<<<cdiv((long)BB * 3 * NN, 256), 256, 0, stream>>>(x, tv, xt);

    // --- x_glo from transformed x ---
    k_xglo<<<cdiv((long)BB * 2 * NN, 256), 256, 0, stream>>>(xt, xg);

    // --- LAB stack 1-4 ---
    lab(xt, 3, 1, l1k, l1v, l1s, 64, x2, wsf, stream);
    lab(x2, 64, 0, l2k, l2v, l2s, 64, x3, wsf, stream);
    lab(x3, 64, 0, l3k, l3v, l3s, 128, x4, wsf, stream);
    lab(x4, 128, 0, l4k, l4v, l4s, 256, x5, wsf, stream);

    // --- GAB ---
    k_concat_xc<<<cdiv((long)BB * 512 * NN, 256), 256, 0, stream>>>(x2, x3, x4, x5, xc);
    float* vals = RA;
    float* f    = RB;
    float* s1   = RB + 17000000;
    float* s2   = RB + 21500000;
    gemm(g1, xc, vals, BB, 2048, 512, NN, 0, stream); bn(vals, st, BB, 2048, NN, 1, stream);
    k_add_bcast<<<cdiv((long)BB * 2048 * NN, 256), 256, 0, stream>>>(vals, xg, f);
    gemm(g2, f, s1, BB, 512, 2048, NN, 0, stream);    bn(s1, st, BB, 512, NN, 1, stream);
    gemm(g3, s1, s2, BB, 128, 512, NN, 0, stream);    bn(s2, st, BB, 128, NN, 1, stream);
    gemm(g4, s2, sn, BB, 1, 128, NN, 0, stream);      bn(sn, st, BB, 1, NN, 1, stream);
    k_softmax_n<<<BB, 256, 0, stream>>>(sn);
    k_gab_gather<<<cdiv((long)BB * 2048 * 32, 256), 256, 0, stream>>>(vals, sn, gbuf);

    // --- c7 on label vector ---
    gemm(c7w, l, lf, BB, 64, 16, 1, 0, stream); bn(lf, st, BB, 64, 1, 1, stream);

    // --- concat + c8/c9/c10 ---
    k_concat_h<<<cdiv((long)BB * 2624 * NN, 256), 256, 0, stream>>>(gbuf, lf, x2, x3, x4, x5, RA);
    float* x8 = RB;
    float* x9 = RB + 8000000;
    gemm(c8w, RA, x8, BB, 512, 2624, NN, 0, stream); bn(x8, st, BB, 512, NN, 1, stream);
    gemm(c9w, x8, x9, BB, 256, 512, NN, 0, stream);  bn(x9, st, BB, 256, NN, 1, stream);
    gemm(c10w, x9, hA, BB, 128, 256, NN, 0, stream); bn(hA, st, BB, 128, NN, 1, stream);

    // --- LAB stack 5-7 (last writes d_out: (B,50,N)) ---
    lab(hA, 128, 0, l5k, l5v, l5s, 64, hB, wsf, stream);
    lab(hB, 64, 0, l6k, l6v, l6s, 64, hC, wsf, stream);
    lab(hC, 64, 0, l7k, l7v, l7s, 50, (float*)d_out, wsf, stream);
}